// OutConv_37898791420065
// MI455X (gfx1250) — compile-verified
//
#include <hip/hip_runtime.h>
#include <hip/hip_bf16.h>

#define NUM_BINS 256
#define IMG_PIX  (224 * 224)   // 50176 pixels per image
#define NIMG     8
#define BPI      49            // blocks per image: 49 * 1024 px = 50176

typedef __attribute__((ext_vector_type(2))) float v2f;
typedef __attribute__((ext_vector_type(8))) float v8f;

// ---------------------------------------------------------------------------
// Kernel 0: zero the histogram counters in workspace (harness poisons ws once
// and never re-zeros between graph replays, so we must do it every call).
// ---------------------------------------------------------------------------
__global__ void zero_cnt_kernel(unsigned int* __restrict__ cnt) {
    int t = blockIdx.x * blockDim.x + threadIdx.x;
    if (t < NIMG * NUM_BINS) cnt[t] = 0u;
}

// ---------------------------------------------------------------------------
// Kernel 1: per-image 256-bin histogram. LDS-local atomics, one global atomic
// per (block, bin) to combine. 392 blocks x 256 threads x 4 px (int4 loads).
// ---------------------------------------------------------------------------
__global__ void hist_kernel(const int* __restrict__ x,
                            unsigned int* __restrict__ cnt) {
    __shared__ unsigned int lds[NUM_BINS];
    const int img   = blockIdx.x / BPI;
    const int chunk = blockIdx.x % BPI;
    const int t     = threadIdx.x;

    lds[t] = 0u;
    __syncthreads();

    const int4 v = ((const int4*)(x + img * IMG_PIX + chunk * 1024))[t];
    atomicAdd(&lds[v.x], 1u);
    atomicAdd(&lds[v.y], 1u);
    atomicAdd(&lds[v.z], 1u);
    atomicAdd(&lds[v.w], 1u);
    __syncthreads();

    const unsigned int c = lds[t];
    if (c) atomicAdd(&cnt[img * NUM_BINS + t], c);
}

// ---------------------------------------------------------------------------
// Kernel 2: gate = sigmoid(hist @ W^T + b) via V_WMMA_F32_16X16X4_F32.
//   M = 16 (batch rows 0..7 valid, 8..15 zero-padded), N = 256, K = 256.
//   1 block x 512 threads = 16 wave32's; wave w owns N-tile [16w, 16w+16).
//   K loop: 64 WMMA steps of K=4.
//
// Fragment layouts (CDNA5 ISA 7.12.2, 32-bit operands, wave32):
//   A (16x4):  lane<16 -> M=lane,    VGPR{0,1} = K{0,1}
//              lane>=16 -> M=lane-16, VGPR{0,1} = K{2,3}
//   B (4x16):  lane<16 -> N=lane,    VGPR{0,1} = K{0,1}
//              lane>=16 -> N=lane-16, VGPR{0,1} = K{2,3}
//   C/D (16x16): VGPR v: lanes 0-15 -> M=v (valid batch rows), N=lane;
//                        lanes 16-31 -> M=v+8 (zero-pad rows, discarded).
// B[k][n] = W[n0+n][k] (because gate = hist @ W^T), so each lane's B frag is a
// contiguous float2 from one row of W -> a single global_load_b64 per step.
// EXEC is all-ones and uniform throughout the WMMA loop (WMMA requirement).
// ---------------------------------------------------------------------------
__global__ void gate_kernel(const unsigned int* __restrict__ cnt,
                            const float* __restrict__ W,     // 256x256 row-major
                            const float* __restrict__ bias,  // 256
                            float* __restrict__ gate) {      // 8x256
    __shared__ float histLds[16][NUM_BINS];   // rows 8..15 are zero padding

    const int t = threadIdx.x;                // 0..511
    const float invN = 1.0f / (float)IMG_PIX;
    for (int idx = t; idx < 16 * NUM_BINS; idx += 512) {
        const int row = idx >> 8;
        const int col = idx & (NUM_BINS - 1);
        histLds[row][col] =
            (row < NIMG) ? (float)cnt[row * NUM_BINS + col] * invN : 0.0f;
    }
    __syncthreads();

    const int wave = t >> 5;                  // 0..15 -> N tile index
    const int lane = t & 31;
    const int n0   = wave * 16;
    const int m    = lane & 15;               // A row / C column-lane
    const int koff = (lane >> 4) << 1;        // 0 for lanes 0-15, 2 for 16-31
    const float* __restrict__ wrow = W + (n0 + m) * NUM_BINS;

    v8f c = {};
    #pragma unroll 4
    for (int k0 = 0; k0 < NUM_BINS; k0 += 4) {
        const int kb = k0 + koff;
        v2f a;
        a.x = histLds[m][kb];
        a.y = histLds[m][kb + 1];
        const v2f b = *(const v2f*)(wrow + kb);    // 8B aligned (kb even)
        // 8 args: (neg_a, A, neg_b, B, c_mod, C, reuse_a, reuse_b)
        c = __builtin_amdgcn_wmma_f32_16x16x4_f32(
                false, a, false, b, (short)0, c, false, false);
    }

    // Lanes 0-15 hold the 8 valid batch rows (M = v for VGPR v = 0..7).
    if (lane < 16) {
        const int n = n0 + lane;
        const float bb = bias[n];
        #pragma unroll
        for (int v = 0; v < 8; ++v) {
            const float z = c[v] + bb;
            gate[v * NUM_BINS + n] = 1.0f / (1.0f + __expf(-z));
        }
    }
}

// ---------------------------------------------------------------------------
// Kernel 3: out[b,h,w] = argmax_ch(onehot * gate), exact semantics:
//   gate[b,v] > 0  -> v
//   gate[b,v] == 0 -> 0            (all channels tie at 0)
//   gate[b,v] < 0  -> (v==0)?1:0   (first zero channel wins)
// Gate row cached in LDS; int4 in / float4 out.
// ---------------------------------------------------------------------------
__global__ void out_kernel(const int* __restrict__ x,
                           const float* __restrict__ gate,
                           float* __restrict__ out) {
    __shared__ float g[NUM_BINS];
    const int img   = blockIdx.x / BPI;
    const int chunk = blockIdx.x % BPI;
    const int t     = threadIdx.x;

    g[t] = gate[img * NUM_BINS + t];
    __syncthreads();

    const int base = img * IMG_PIX + chunk * 1024;
    const int4 v = ((const int4*)(x + base))[t];

    auto pick = [&](int vv) -> float {
        const float gv = g[vv];
        if (gv > 0.0f) return (float)vv;
        if (gv == 0.0f) return 0.0f;
        return (vv == 0) ? 1.0f : 0.0f;
    };

    float4 r;
    r.x = pick(v.x);
    r.y = pick(v.y);
    r.z = pick(v.z);
    r.w = pick(v.w);
    ((float4*)(out + base))[t] = r;
}

// ---------------------------------------------------------------------------
extern "C" void kernel_launch(void* const* d_in, const int* in_sizes, int n_in,
                              void* d_out, int out_size, void* d_ws, size_t ws_size,
                              hipStream_t stream) {
    const int*   x    = (const int*)d_in[0];     // (8,1,224,224) int32
    const float* W    = (const float*)d_in[1];   // (256,256) f32
    const float* bias = (const float*)d_in[2];   // (256,) f32
    float*       out  = (float*)d_out;           // (8,1,224,224) f32

    unsigned int* cnt  = (unsigned int*)d_ws;                       // 8*256 u32
    float*        gate = (float*)((char*)d_ws +
                                  NIMG * NUM_BINS * sizeof(unsigned int)); // 8*256 f32

    zero_cnt_kernel<<<(NIMG * NUM_BINS + 255) / 256, 256, 0, stream>>>(cnt);
    hist_kernel<<<NIMG * BPI, 256, 0, stream>>>(x, cnt);
    gate_kernel<<<1, 512, 0, stream>>>(cnt, W, bias, gate);
    out_kernel<<<NIMG * BPI, 256, 0, stream>>>(x, gate, out);
}